// ShiftReduceSequenceEmbedder_37520834298033
// MI455X (gfx1250) — compile-verified
//
#include <hip/hip_runtime.h>

// Problem constants (from reference)
constexpr int kB    = 16;
constexpr int kL    = 2048;
constexpr int kS    = 1024;
constexpr int kDE   = 512;
constexpr int kD    = 256;
constexpr int kNSIL = 8;
constexpr int kA    = 64;

// GEMM tiling
constexpr int KCH    = 16;    // K rows per TDM chunk
constexpr int LPITCH = 264;   // LDS row pitch in dwords (256 + 8 TDM pad)

typedef __attribute__((ext_vector_type(2))) float        v2f;
typedef __attribute__((ext_vector_type(8))) float        v8f;
typedef __attribute__((ext_vector_type(4))) unsigned int u32x4;
typedef __attribute__((ext_vector_type(8))) unsigned int u32x8;

// ---------------------------------------------------------------------------
// Kernel 0: reset compaction counter (ws[0]); list lives at ws[16..]
// ---------------------------------------------------------------------------
__global__ void zero_counter(int* __restrict__ ws) {
    if (threadIdx.x == 0) ws[0] = 0;
}

// ---------------------------------------------------------------------------
// Kernel 1: resolve type-0/1 rows (pure gather, memory-bound),
//           compact type-2 rows into a worklist for the WMMA GEMM.
// ---------------------------------------------------------------------------
__global__ __launch_bounds__(256) void gather_select(
    const float* __restrict__ stack_nodes,   // (NN, D)
    const float* __restrict__ silent,        // (NSIL, D)
    const float* __restrict__ op_emb,        // (VOP, D)
    const int*   __restrict__ op_tokens,     // (B, L)
    const int*   __restrict__ arg_idx,       // (B, L)
    const int*   __restrict__ item_type,     // (B, L)
    const int*   __restrict__ avail,         // (B, L, A)
    float*       __restrict__ out,           // (B, L, D)
    int*         __restrict__ ws)
{
    const int wave = threadIdx.x >> 5;
    const int lane = threadIdx.x & 31;
    const int row  = blockIdx.x * 8 + wave;          // 0 .. B*L-1

    const int t = item_type[row];                    // uniform per wave
    if (t == 2) {
        if (lane == 0) {
            int p = atomicAdd(&ws[0], 1);
            ws[16 + p] = row;
        }
        return;                                      // GEMM pass writes this row
    }

    const float* src;
    if (t == 0) {
        src = op_emb + (size_t)op_tokens[row] * kD;
    } else {                                         // t == 1
        int a = arg_idx[row];
        if (a < kNSIL) {
            src = silent + (size_t)a * kD;
        } else {
            int pos = a - kNSIL;
            if (pos > kA - 1) pos = kA - 1;
            int node = avail[(size_t)row * kA + pos];
            src = stack_nodes + (size_t)node * kD;
        }
    }

    const float4* s4 = (const float4*)src + lane * 2;
    float4*       o4 = (float4*)(out + (size_t)row * kD) + lane * 2;
    o4[0] = s4[0];
    o4[1] = s4[1];
}

// ---------------------------------------------------------------------------
// TDM: DMA one (KCH x 256) fp32 tile of w_shift into LDS, row-major with an
// 8-dword pad every 256 dwords (-> 264-dword pitch, bank-conflict-free reads).
// 2-D descriptor, groups 2/3 NULL. Tracked on TENSORcnt.
// ---------------------------------------------------------------------------
__device__ __forceinline__ void tdm_load_chunk(const float* gtile, unsigned lds_byte_addr) {
    const unsigned long long ga = (unsigned long long)(uintptr_t)gtile;
    u32x4 g0;
    g0[0] = 1u;                                        // count=1, user load descriptor
    g0[1] = lds_byte_addr;                             // LDS byte address
    g0[2] = (unsigned)ga;                              // global_addr[31:0] (tile start)
    g0[3] = (unsigned)((ga >> 32) & 0x01FFFFFFu)       // global_addr[56:32]
          | (2u << 30);                                // type = 2 ("image")
    u32x8 g1;
    g1[0] = (2u << 16)                                 // data_size = 4 bytes
          | (1u << 20)                                 // pad_enable
          | (7u << 22)                                 // pad_interval: every 256 dwords
          | (7u << 25);                                // pad_amount: 8 dwords
    g1[1] = ((unsigned)kD)  << 16;                     // tensor_dim0[15:0] = 256
    g1[2] = ((unsigned)kDE) << 16;                     // tensor_dim1[15:0] = 512
    g1[3] = ((unsigned)kD)  << 16;                     // tile_dim0 = 256
    g1[4] = (unsigned)KCH;                             // tile_dim1 = 16, tile_dim2 = 0
    g1[5] = (unsigned)kD;                              // tensor_dim0_stride[31:0] = 256
    g1[6] = 0u;
    g1[7] = 0u;
    asm volatile("tensor_load_to_lds %0, %1" :: "s"(g0), "s"(g1) : "memory");
}

// ---------------------------------------------------------------------------
// Kernel 2: fp32 WMMA GEMM over the compacted type-2 rows.
//   emb_shift[row] = enc_sent[b, shift_idx[b,l], :] @ w_shift + b_shift
// 8 waves/WG; each wave owns one 16-row tile x all 256 columns.
// w_shift chunks stream via double-buffered TDM, overlapped with WMMA.
// ---------------------------------------------------------------------------
__global__ __launch_bounds__(256) void shift_gemm(
    const float* __restrict__ enc_sent,   // (B, S, DE)
    const float* __restrict__ w_shift,    // (DE, D)
    const float* __restrict__ b_shift,    // (D,)
    const int*   __restrict__ shift_idx,  // (B, L)
    float*       __restrict__ out,        // (B, L, D)
    const int*   __restrict__ ws)
{
    __shared__ float Bt[2][KCH * LPITCH];            // 2 x 16.5 KiB, row-major padded

    const int count = ws[0];
    if ((int)blockIdx.x * 128 >= count) return;      // uniform early-out
    const int* list = ws + 16;

    const int tid  = threadIdx.x;
    const int wave = tid >> 5;
    const int lane = tid & 31;
    const int m15  = lane & 15;                      // row in tile / col in N-tile
    const int kb   = (lane >> 4) << 1;               // K lane-half offset: 0 or 2

    // Row for this lane's A operand (clamped -> duplicate identical work)
    const int tile_base = blockIdx.x * 128 + wave * 16;
    int li = tile_base + m15;
    if (li > count - 1) li = count - 1;
    const int myrid = list[li];
    const int bb    = myrid >> 11;                   // row / L
    const int sidx  = shift_idx[myrid];
    const float* aptr = enc_sent + ((size_t)bb * kS + (size_t)sidx) * kDE;

    const unsigned lds_base = (unsigned)(uintptr_t)&Bt[0][0];
    constexpr unsigned kBufBytes = (unsigned)(KCH * LPITCH * 4);

    // Accumulators: 16 N-tiles; fold bias into init (same col for all 8 M rows)
    v8f acc[16];
#pragma unroll
    for (int ct = 0; ct < 16; ++ct) {
        float bias = b_shift[ct * 16 + m15];
        acc[ct] = (v8f){bias, bias, bias, bias, bias, bias, bias, bias};
    }

    constexpr int NCH = kDE / KCH;                   // 32 chunks
    // Prologue: stage chunk 0 into buffer 0
    if (wave == 0) {
        tdm_load_chunk(w_shift, lds_base);
        __builtin_amdgcn_s_wait_tensorcnt(0);
    }
    __syncthreads();

    for (int chunk = 0; chunk < NCH; ++chunk) {
        const int buf = chunk & 1;
        // Prefetch next chunk into the other buffer (runs under the WMMAs)
        if (wave == 0 && chunk + 1 < NCH) {
            tdm_load_chunk(w_shift + (size_t)(chunk + 1) * KCH * kD,
                           lds_base + (unsigned)(buf ^ 1) * kBufBytes);
        }

        const float* bbase = &Bt[buf][0];
        const int    k0    = chunk * KCH;
#pragma unroll
        for (int kk = 0; kk < KCH; kk += 4) {
            // A: 16x4 fp32 tile (lanes 0-15: K=k..k+1, lanes 16-31: K=k+2..k+3)
            v2f a = *(const v2f*)(aptr + (k0 + kk + kb));
#pragma unroll
            for (int ct = 0; ct < 16; ++ct) {
                const int idx = (kk + kb) * LPITCH + ct * 16 + m15;
                v2f b;
                b.x = bbase[idx];                     // B[K][N]
                b.y = bbase[idx + LPITCH];            // B[K+1][N]
                acc[ct] = __builtin_amdgcn_wmma_f32_16x16x4_f32(
                    /*neg_a=*/false, a, /*neg_b=*/false, b,
                    /*c_mod=*/(short)0, acc[ct],
                    /*reuse_a=*/false, /*reuse_b=*/false);
            }
        }

        // Drain this wave's outstanding TDM, then block-wide handoff
        if (wave == 0) __builtin_amdgcn_s_wait_tensorcnt(0);
        __syncthreads();
    }

    // Epilogue: D layout -> lane (v, hi) holds (M = v + hi*8, N = ct*16 + m15)
    const int hi = lane >> 4;
#pragma unroll
    for (int v = 0; v < 8; ++v) {
        int m   = v + hi * 8;
        int rid = __shfl(myrid, m, 32);              // row id held by lane m (0..15)
        float* orow = out + (size_t)rid * kD;
#pragma unroll
        for (int ct = 0; ct < 16; ++ct) {
            orow[ct * 16 + m15] = acc[ct][v];
        }
    }
}

// ---------------------------------------------------------------------------
extern "C" void kernel_launch(void* const* d_in, const int* in_sizes, int n_in,
                              void* d_out, int out_size, void* d_ws, size_t ws_size,
                              hipStream_t stream) {
    const float* enc_sent    = (const float*)d_in[0];
    const float* stack_nodes = (const float*)d_in[1];
    const float* silent      = (const float*)d_in[2];
    const float* op_emb      = (const float*)d_in[3];
    const float* w_shift     = (const float*)d_in[4];
    const float* b_shift     = (const float*)d_in[5];
    const int*   op_tokens   = (const int*)d_in[6];
    const int*   arg_idx     = (const int*)d_in[7];
    const int*   shift_idx   = (const int*)d_in[8];
    const int*   item_type   = (const int*)d_in[9];
    const int*   avail       = (const int*)d_in[10];
    float*       out         = (float*)d_out;
    int*         ws          = (int*)d_ws;

    hipLaunchKernelGGL(zero_counter, dim3(1), dim3(32), 0, stream, ws);
    hipLaunchKernelGGL(gather_select, dim3((kB * kL) / 8), dim3(256), 0, stream,
                       stack_nodes, silent, op_emb, op_tokens, arg_idx,
                       item_type, avail, out, ws);
    hipLaunchKernelGGL(shift_gemm, dim3((kB * kL) / 128), dim3(256), 0, stream,
                       enc_sent, w_shift, b_shift, shift_idx, out, ws);
}